// LiteMLA_21741124452913
// MI455X (gfx1250) — compile-verified
//
#include <hip/hip_runtime.h>

typedef __attribute__((ext_vector_type(2))) float v2f;
typedef __attribute__((ext_vector_type(8))) float v8f;

// ---------------------------------------------------------------------------
// Wave-level fp32 WMMA 16x16 tile: D += A(16xK) * B(Kx16), K multiple of 4.
// A layout (ISA 7.12.2, 32-bit A 16x4): lanes 0-15 M=lane, VGPR0=K0 VGPR1=K1;
// lanes 16-31 M=lane-16, VGPR0=K2 VGPR1=K3.  B mirrors (rows across lanes).
// C/D: VGPR v -> (M=v, N=lane) for lane<16, (M=v+8, N=lane-16) otherwise.
// ---------------------------------------------------------------------------
template <typename FA, typename FB>
__device__ __forceinline__ v8f wmma_loop(int K, FA la, FB lb) {
  const int lane = threadIdx.x & 31;
  const int kh = (lane >> 4) << 1;           // 0 or 2
  v8f acc = {0.f, 0.f, 0.f, 0.f, 0.f, 0.f, 0.f, 0.f};
  for (int k0 = 0; k0 < K; k0 += 4) {
    const int k = k0 + kh;
    v2f a, b;
    a.x = la(k);     a.y = la(k + 1);
    b.x = lb(k);     b.y = lb(k + 1);
    acc = __builtin_amdgcn_wmma_f32_16x16x4_f32(false, a, false, b,
                                                (short)0, acc, false, false);
  }
  return acc;
}

template <typename FS>
__device__ __forceinline__ void wmma_store16(const v8f& acc, FS st) {
  const int lane = threadIdx.x & 31;
  const int col = lane & 15;
  const int mh  = (lane >> 4) * 8;
#pragma unroll
  for (int v = 0; v < 8; ++v) st(mh + v, col, acc[v]);
}

// ---------------------------------------------------------------------------
// Generic batched GEMM: C[prob] = A[prob%aMod] (MxK, lda) * B[prob] (KxN, ldb)
// One wave per 16x16 output tile.
// ---------------------------------------------------------------------------
__global__ __launch_bounds__(256) void k_gemm(
    const float* __restrict__ A, int aMod, long long aStride, int lda,
    const float* __restrict__ B, long long bStride, int ldb,
    float* __restrict__ C, long long cStride, int ldc,
    int K, int nMt, int nNt, int nProb)
{
  const int wave = threadIdx.x >> 5;
  const long long gid = (long long)blockIdx.x * (blockDim.x >> 5) + wave;
  const long long total = (long long)nProb * nMt * nNt;
  if (gid >= total) return;
  const int tilesPer = nMt * nNt;
  const int prob = (int)(gid / tilesPer);
  const int t    = (int)(gid % tilesPer);
  const int mt = t / nNt, nt = t % nNt;
  const float* Ap = A + (long long)(prob % aMod) * aStride;
  const float* Bp = B + (long long)prob * bStride;
  float*       Cp = C + (long long)prob * cStride;
  const int lm = threadIdx.x & 15;
  const int rowA = mt * 16 + lm;
  const int colB = nt * 16 + lm;
  v8f acc = wmma_loop(K,
      [&](int k) { return Ap[(long long)rowA * lda + k]; },
      [&](int k) { return Bp[(long long)k * ldb + colB]; });
  wmma_store16(acc, [&](int r, int c, float v) {
      Cp[(long long)(mt * 16 + r) * ldc + nt * 16 + c] = v;
  });
}

// ---------------------------------------------------------------------------
// Depthwise 5x5 conv, pad 2, on (B,256,128,128)
// ---------------------------------------------------------------------------
__global__ __launch_bounds__(256) void k_dw_img(
    const float* __restrict__ q, const float* __restrict__ w,
    float* __restrict__ qdw)
{
  const int idx = blockIdx.x * blockDim.x + threadIdx.x;   // 8,388,608
  const int x = idx & 127, y = (idx >> 7) & 127;
  const int c = (idx >> 14) & 255;
  const float* wp = w + c * 25;
  const long long base = (long long)(idx >> 14) << 14;     // (b*256+c)*16384
  float acc = 0.f;
#pragma unroll
  for (int dy = 0; dy < 5; ++dy) {
    const int yy = y + dy - 2;
    if ((unsigned)yy >= 128u) continue;
#pragma unroll
    for (int dx = 0; dx < 5; ++dx) {
      const int xx = x + dx - 2;
      if ((unsigned)xx >= 128u) continue;
      acc += wp[dy * 5 + dx] * q[base + yy * 128 + xx];
    }
  }
  qdw[idx] = acc;
}

// ---------------------------------------------------------------------------
// Depthwise 5x5x5 conv, pad 2, on (B,512,32,32,32)
// ---------------------------------------------------------------------------
__global__ __launch_bounds__(256) void k_dw_vox(
    const float* __restrict__ kv, const float* __restrict__ w,
    float* __restrict__ kvdw)
{
  const int idx = blockIdx.x * blockDim.x + threadIdx.x;   // 33,554,432
  const int x = idx & 31, y = (idx >> 5) & 31, z = (idx >> 10) & 31;
  const int c = (idx >> 15) & 511;
  const float* wp = w + c * 125;
  const long long base = (long long)(idx >> 15) << 15;     // (b*512+c)*32768
  float acc = 0.f;
  for (int dz = 0; dz < 5; ++dz) {
    const int zz = z + dz - 2;
    if ((unsigned)zz >= 32u) continue;
    for (int dy = 0; dy < 5; ++dy) {
      const int yy = y + dy - 2;
      if ((unsigned)yy >= 32u) continue;
      for (int dx = 0; dx < 5; ++dx) {
        const int xx = x + dx - 2;
        if ((unsigned)xx >= 32u) continue;
        acc += wp[dz * 25 + dy * 5 + dx] * kv[base + zz * 1024 + yy * 32 + xx];
      }
    }
  }
  kvdw[idx] = acc;
}

// ---------------------------------------------------------------------------
// Patch conv img: stride-2 2x2, 512->512, reads concat[q,qs] via select.
// GEMM: M=4096 tokens, N=512 out-ch, K=2048. Output token-major qt (B,4096,512).
// ---------------------------------------------------------------------------
__global__ __launch_bounds__(256) void k_patch_img(
    const float* __restrict__ q, const float* __restrict__ qs,
    const float* __restrict__ W, const float* __restrict__ bias,
    float* __restrict__ qt)
{
  const int wave = threadIdx.x >> 5;
  const int gid = blockIdx.x * (blockDim.x >> 5) + wave;   // 16384 waves
  const int b = gid >> 13;                                  // 8192 tiles/batch
  const int t = gid & 8191;
  const int mt = t >> 5, nt = t & 31;
  const int lm = threadIdx.x & 15;
  const int token = mt * 16 + lm;
  const int r0 = (token >> 6) * 2, c0 = (token & 63) * 2;
  const long long qbase = (long long)b * 256 * 16384;
  const int co = nt * 16 + lm;
  const float* Wc = W + (long long)co * 2048;
  v8f acc = wmma_loop(2048,
      [&](int k) {
        const int ci = k >> 2, ky = (k >> 1) & 1, kx = k & 1;
        const float* src = (ci < 256) ? q : qs;
        return src[qbase + (long long)(ci & 255) * 16384 + (r0 + ky) * 128 + c0 + kx];
      },
      [&](int k) { return Wc[k]; });
  wmma_store16(acc, [&](int r, int c, float v) {
      const int tok = mt * 16 + r, o = nt * 16 + c;
      qt[((long long)b * 4096 + tok) * 512 + o] = v + bias[o];
  });
}

// ---------------------------------------------------------------------------
// Patch conv vox: stride-2 2x2x2, 1024->1024. GEMM M=4096, N=1024, K=8192.
// Output token-major kvt (B,4096,1024).  This is the 137-GFLOP hot loop.
// ---------------------------------------------------------------------------
__global__ __launch_bounds__(256) void k_patch_vox(
    const float* __restrict__ kv, const float* __restrict__ kvs,
    const float* __restrict__ W, const float* __restrict__ bias,
    float* __restrict__ kvt)
{
  const int wave = threadIdx.x >> 5;
  const int gid = blockIdx.x * (blockDim.x >> 5) + wave;   // 32768 waves
  const int b = gid >> 14;                                  // 16384 tiles/batch
  const int t = gid & 16383;
  const int mt = t >> 6, nt = t & 63;
  const int lm = threadIdx.x & 15;
  const int token = mt * 16 + lm;
  const int z0 = (token >> 8) * 2, y0 = ((token >> 4) & 15) * 2, x0 = (token & 15) * 2;
  const long long vbase = (long long)b * 512 * 32768;
  const int co = nt * 16 + lm;
  const float* Wc = W + (long long)co * 8192;
  v8f acc = wmma_loop(8192,
      [&](int k) {
        const int ci = k >> 3, r = k & 7;
        const int kz = r >> 2, ky = (r >> 1) & 1, kx = r & 1;
        const float* src = (ci < 512) ? kv : kvs;
        return src[vbase + (long long)(ci & 511) * 32768 +
                   (z0 + kz) * 1024 + (y0 + ky) * 32 + x0 + kx];
      },
      [&](int k) { return Wc[k]; });
  wmma_store16(acc, [&](int r, int c, float v) {
      const int tok = mt * 16 + r, o = nt * 16 + c;
      kvt[((long long)b * 4096 + tok) * 1024 + o] = v + bias[o];
  });
}

// ---------------------------------------------------------------------------
// ctx[bg] = ReLU(K)^T (32x4096) @ [V | 1] (4096x33), padded to 48 cols (zeros).
// ---------------------------------------------------------------------------
__global__ __launch_bounds__(256) void k_ctx(
    const float* __restrict__ kvt, float* __restrict__ ctx)
{
  const int wave = threadIdx.x >> 5;
  const int gid = blockIdx.x * (blockDim.x >> 5) + wave;   // 192 waves
  if (gid >= 32 * 6) return;
  const int bg = gid / 6, t = gid % 6;
  const int mt = t / 3, nt = t % 3;
  const int lm = threadIdx.x & 15;
  const float* base = kvt + (long long)bg * 262144;        // (B*16) groups of 4096x64
  const int am = mt * 16 + lm;                             // k-dim index (<32)
  const int j  = nt * 16 + lm;                             // v col (<48)
  v8f acc = wmma_loop(4096,
      [&](int k) { float x = base[(long long)k * 64 + am]; return x > 0.f ? x : 0.f; },
      [&](int k) {
        if (j < 32) return base[(long long)k * 64 + 32 + j];
        return (j == 32) ? 1.0f : 0.0f;
      });
  wmma_store16(acc, [&](int r, int c, float v) {
      ctx[(long long)bg * 1536 + (mt * 16 + r) * 48 + nt * 16 + c] = v;
  });
}

// ---------------------------------------------------------------------------
// out[bg] = ReLU(Q) (4096x32) @ ctx (32x48); store only cols 0..32 (stride 33).
// ---------------------------------------------------------------------------
__global__ __launch_bounds__(256) void k_attn_out(
    const float* __restrict__ qt, const float* __restrict__ ctx,
    float* __restrict__ outn)
{
  const int wave = threadIdx.x >> 5;
  const int gid = blockIdx.x * (blockDim.x >> 5) + wave;   // 24576 waves
  const int bg = gid / 768, t = gid % 768;
  const int mt = t / 3, nt = t % 3;
  const int lm = threadIdx.x & 15;
  const float* qb = qt + (long long)bg * 131072;           // 4096 rows x 32
  const float* cb = ctx + (long long)bg * 1536;
  const int token = mt * 16 + lm;
  const int j = nt * 16 + lm;
  v8f acc = wmma_loop(32,
      [&](int k) { float x = qb[(long long)token * 32 + k]; return x > 0.f ? x : 0.f; },
      [&](int k) { return cb[k * 48 + j]; });
  wmma_store16(acc, [&](int r, int c, float v) {
      const int tok = mt * 16 + r, jj = nt * 16 + c;
      if (jj < 33) outn[(long long)bg * 135168 + (long long)tok * 33 + jj] = v;
  });
}

__global__ __launch_bounds__(256) void k_recip(
    const float* __restrict__ outn, float* __restrict__ rnorm)
{
  const int idx = blockIdx.x * blockDim.x + threadIdx.x;   // 131072
  rnorm[idx] = 1.0f / (outn[(long long)idx * 33 + 32] + 1e-15f);
}

// ---------------------------------------------------------------------------
// lin[bg] = normalized-out (4096x32) @ w_lin^T (32x128)
// ---------------------------------------------------------------------------
__global__ __launch_bounds__(256) void k_lin(
    const float* __restrict__ outn, const float* __restrict__ rnorm,
    const float* __restrict__ wlin, float* __restrict__ lin)
{
  const int wave = threadIdx.x >> 5;
  const int gid = blockIdx.x * (blockDim.x >> 5) + wave;   // 65536 waves
  const int bg = gid >> 11, t = gid & 2047;
  const int mt = t >> 3, nt = t & 7;
  const int lm = threadIdx.x & 15;
  const float* ob = outn + (long long)bg * 135168;
  const int token = mt * 16 + lm;
  const float rn = rnorm[bg * 4096 + token];
  const int o = nt * 16 + lm;
  v8f acc = wmma_loop(32,
      [&](int k) { return ob[(long long)token * 33 + k] * rn; },
      [&](int k) { return wlin[o * 32 + k]; });
  wmma_store16(acc, [&](int r, int c, float v) {
      lin[(long long)bg * 524288 + (long long)(mt * 16 + r) * 128 + nt * 16 + c] = v;
  });
}

// ---------------------------------------------------------------------------
// 1x1 proj over the swap/reshape view: p[b,oc,pix] = sum_ch w_proj[oc,ch]*X.
// X[b,ch,pix] = lin[b, ch/32, pix%4096, (ch%32)*4 + pix/4096]
// ---------------------------------------------------------------------------
__global__ __launch_bounds__(256) void k_proj(
    const float* __restrict__ wproj, const float* __restrict__ lin,
    float* __restrict__ p)
{
  const int wave = threadIdx.x >> 5;
  const int gid = blockIdx.x * (blockDim.x >> 5) + wave;   // 4096 waves
  const int b = gid >> 11, t = gid & 2047;
  const int mt = t >> 10, nt = t & 1023;
  const int lm = threadIdx.x & 15;
  const int oc = mt * 16 + lm;
  const int pix = nt * 16 + lm;
  const int n = pix & 4095, ph = pix >> 12;
  const float* lb = lin + (long long)b * 8388608;
  v8f acc = wmma_loop(512,
      [&](int k) { return wproj[oc * 512 + k]; },
      [&](int k) {
        const int g = k >> 5;
        const int o = ((k & 31) << 2) + ph;
        return lb[(long long)g * 524288 + (long long)n * 128 + o];
      });
  wmma_store16(acc, [&](int r, int c, float v) {
      p[(long long)b * 524288 + (long long)(mt * 16 + r) * 16384 + nt * 16 + c] = v;
  });
}

// ---------------------------------------------------------------------------
// BatchNorm stats per channel over (B, H, W) = 32768 elements.
// ---------------------------------------------------------------------------
__global__ __launch_bounds__(256) void k_bnstats(
    const float* __restrict__ p, float* __restrict__ stats)
{
  __shared__ float s1[256], s2[256];
  const int c = blockIdx.x, tid = threadIdx.x;
  float s = 0.f, ss = 0.f;
  for (int i = tid; i < 32768; i += 256) {
    const int b = i >> 14, pix = i & 16383;
    const float v = p[(long long)b * 524288 + c * 16384 + pix];
    s += v; ss += v * v;
  }
  s1[tid] = s; s2[tid] = ss;
  __syncthreads();
  for (int off = 128; off > 0; off >>= 1) {
    if (tid < off) { s1[tid] += s1[tid + off]; s2[tid] += s2[tid + off]; }
    __syncthreads();
  }
  if (tid == 0) {
    const float mean = s1[0] * (1.f / 32768.f);
    stats[c] = mean;
    stats[32 + c] = s2[0] * (1.f / 32768.f) - mean * mean;
  }
}

__global__ __launch_bounds__(256) void k_final(
    const float* __restrict__ x, const float* __restrict__ p,
    const float* __restrict__ stats, const float* __restrict__ gamma,
    const float* __restrict__ beta, float* __restrict__ out)
{
  const int idx = blockIdx.x * blockDim.x + threadIdx.x;   // 1,048,576
  const int c = (idx >> 14) & 31;
  const float m = stats[c], v = stats[32 + c];
  out[idx] = x[idx] + (p[idx] - m) * rsqrtf(v + 1e-6f) * gamma[c] + beta[c];
}

// ---------------------------------------------------------------------------
extern "C" void kernel_launch(void* const* d_in, const int* in_sizes, int n_in,
                              void* d_out, int out_size, void* d_ws, size_t ws_size,
                              hipStream_t stream) {
  const float* x        = (const float*)d_in[0];
  const float* y        = (const float*)d_in[1];
  const float* w_qkv_i  = (const float*)d_in[2];
  const float* w_qkv_v  = (const float*)d_in[3];
  const float* w_dw_i   = (const float*)d_in[4];
  const float* w_pw_i   = (const float*)d_in[5];
  const float* w_dw_v   = (const float*)d_in[6];
  const float* w_pw_v   = (const float*)d_in[7];
  const float* w_pat_i  = (const float*)d_in[8];
  const float* b_pat_i  = (const float*)d_in[9];
  const float* w_pat_v  = (const float*)d_in[10];
  const float* b_pat_v  = (const float*)d_in[11];
  const float* w_lin    = (const float*)d_in[12];
  const float* w_proj   = (const float*)d_in[13];
  const float* gamma    = (const float*)d_in[14];
  const float* beta     = (const float*)d_in[15];

  float* ws = (float*)d_ws;
  float* q     = ws;                 //  8,388,608
  float* qs    = ws + 8388608LL;     //  8,388,608
  float* kv    = ws + 16777216LL;    // 33,554,432
  float* kvs   = ws + 50331648LL;    // 33,554,432
  float* qdw   = ws + 83886080LL;    //  8,388,608 (dead after pw_img -> reused as outn)
  float* kvdw  = ws + 92274688LL;    // 33,554,432 (dead after pw_vox -> reused as lin+rnorm)
  float* qt    = ws + 125829120LL;   //  4,194,304
  float* kvt   = ws + 130023424LL;   //  8,388,608
  float* ctx   = ws + 138412032LL;   //     49,152
  float* p     = ws + 138461184LL;   //  1,048,576
  float* stats = ws + 139509760LL;   //         64
  float* outn  = qdw;                //  4,325,376 (fits in qdw region)
  float* lin   = kvdw;               // 16,777,216
  float* rnorm = kvdw + 16777216LL;  //    131,072

  const dim3 blk(256);

  // 1x1 convs (GEMM, K=32): q = Wq @ x ; kv = Wkv @ y
  k_gemm<<<4096,  blk, 0, stream>>>(w_qkv_i, 1, 0, 32, x, 524288LL, 16384,
                                    q, 4194304LL, 16384, 32, 16, 1024, 2);
  k_gemm<<<16384, blk, 0, stream>>>(w_qkv_v, 1, 0, 32, y, 1048576LL, 32768,
                                    kv, 16777216LL, 32768, 32, 32, 2048, 2);
  // depthwise + grouped pointwise (multi-scale branch)
  k_dw_img<<<32768, blk, 0, stream>>>(q, w_dw_i, qdw);
  k_gemm<<<4096,  blk, 0, stream>>>(w_pw_i, 8, 1024LL, 32, qdw, 524288LL, 16384,
                                    qs, 524288LL, 16384, 32, 2, 1024, 16);
  k_dw_vox<<<131072, blk, 0, stream>>>(kv, w_dw_v, kvdw);
  k_gemm<<<16384, blk, 0, stream>>>(w_pw_v, 16, 1024LL, 32, kvdw, 1048576LL, 32768,
                                    kvs, 1048576LL, 32768, 32, 2, 2048, 32);
  // patch embeds (stride-2 convs as im2col GEMMs), token-major outputs
  k_patch_img<<<2048, blk, 0, stream>>>(q, qs, w_pat_i, b_pat_i, qt);
  k_patch_vox<<<4096, blk, 0, stream>>>(kv, kvs, w_pat_v, b_pat_v, kvt);
  // linear attention
  k_ctx<<<24, blk, 0, stream>>>(kvt, ctx);
  k_attn_out<<<3072, blk, 0, stream>>>(qt, ctx, outn);
  k_recip<<<512, blk, 0, stream>>>(outn, rnorm);
  k_lin<<<8192, blk, 0, stream>>>(outn, rnorm, w_lin, lin);
  // projection + batchnorm + residual
  k_proj<<<512, blk, 0, stream>>>(w_proj, lin, p);
  k_bnstats<<<32, blk, 0, stream>>>(p, stats);
  k_final<<<4096, blk, 0, stream>>>(x, p, stats, gamma, beta, (float*)d_out);
}